// GAT_5342939316742
// MI455X (gfx1250) — compile-verified
//
#include <hip/hip_runtime.h>
#include <hip/hip_bf16.h>
#include <math.h>

// Problem constants (match reference)
#define NNODES 50000
#define NEDGES 800000
#define INC    256
#define HIDC   256
#define OUTC   64
#define HEADS1 4
#define NEG_SLOPE 0.2f

typedef float v2f __attribute__((ext_vector_type(2)));
typedef float v8f __attribute__((ext_vector_type(8)));
typedef int   v4i __attribute__((vector_size(16)));

#if defined(__has_builtin)
#if __has_builtin(__builtin_amdgcn_global_load_async_to_lds_b128)
#define HAVE_ASYNC_LDS 1
#endif
#endif

__device__ __forceinline__ float leaky(float x) { return x > 0.f ? x : NEG_SLOPE * x; }

// Order-independent float atomic max via int/uint monotonic mapping.
__device__ __forceinline__ void atomicMaxF(float* addr, float v) {
    if (v >= 0.f) atomicMax((int*)addr, __float_as_int(v));
    else          atomicMin((unsigned int*)addr, __float_as_uint(v));
}

__global__ void fill_kernel(float* __restrict__ p, float v, int n) {
    int i = blockIdx.x * blockDim.x + threadIdx.x;
    if (i < n) p[i] = v;
}

// C[M x Nc] = A[M x K] * B[K x Nc], f32 WMMA 16x16x4.
// Block = 256 threads = 8 waves; the whole block shares one K x 64 B-panel
// staged in LDS via gfx1250 async global->LDS copies; each wave computes a
// 16x64 output tile (4 accumulators, A fragment reused).
// K must be a multiple of 4 and <= 256 (LDS panel is 256*64 floats = 64 KB).
__global__ void gemm_wmma_f32(const float* __restrict__ A, const float* __restrict__ B,
                              float* __restrict__ C, int M, int K, int Nc) {
    __shared__ float Bs[256 * 64];  // [k][n] row-major panel
    const int t     = threadIdx.x;
    const int lane  = t & 31;
    const int wave  = t >> 5;
    const int half  = lane >> 4;
    const int lm    = lane & 15;
    const int nBase = blockIdx.y * 64;

    // ---- cooperative panel stage: K*64 floats, 16B chunks, no divergence ----
    const int nChunks = (K * 64) >> 2;  // 4096 for K=256
    for (int cid = t; cid < nChunks; cid += 256) {
        int f  = cid << 2;       // first float of this 16B chunk
        int k  = f >> 6;
        int nn = f & 63;
        const float* gsrc = B + (size_t)k * Nc + nBase + nn;
        float*       ldst = &Bs[f];
#ifdef HAVE_ASYNC_LDS
        __builtin_amdgcn_global_load_async_to_lds_b128(
            (v4i*)const_cast<float*>(gsrc), (v4i*)ldst, 0, 0);
#else
        ldst[0] = gsrc[0]; ldst[1] = gsrc[1]; ldst[2] = gsrc[2]; ldst[3] = gsrc[3];
#endif
    }
#ifdef HAVE_ASYNC_LDS
    asm volatile("s_wait_asynccnt 0x0" ::: "memory");
#endif
    __syncthreads();

    const int mTile = blockIdx.x * 8 + wave;
    const int mBase = mTile * 16;
    if (mBase >= M) return;

    v8f acc0{}, acc1{}, acc2{}, acc3{};
    const float* arow = A + (size_t)(mBase + lm) * K;
    for (int k0 = 0; k0 < K; k0 += 4) {
        const int ka = k0 + half * 2;
        v2f a; a.x = arow[ka]; a.y = arow[ka + 1];
        const float* bp = &Bs[ka * 64 + lm];
        v2f b0; b0.x = bp[0];  b0.y = bp[64 + 0];
        v2f b1; b1.x = bp[16]; b1.y = bp[64 + 16];
        v2f b2; b2.x = bp[32]; b2.y = bp[64 + 32];
        v2f b3; b3.x = bp[48]; b3.y = bp[64 + 48];
        acc0 = __builtin_amdgcn_wmma_f32_16x16x4_f32(false, a, false, b0, (short)0, acc0, false, false);
        acc1 = __builtin_amdgcn_wmma_f32_16x16x4_f32(false, a, false, b1, (short)0, acc1, false, false);
        acc2 = __builtin_amdgcn_wmma_f32_16x16x4_f32(false, a, false, b2, (short)0, acc2, false, false);
        acc3 = __builtin_amdgcn_wmma_f32_16x16x4_f32(false, a, false, b3, (short)0, acc3, false, false);
    }
    // C/D layout: VGPR v -> row v (lanes 0-15) or v+8 (lanes 16-31), col = lane%16
    for (int v = 0; v < 8; ++v) {
        int row = mBase + v + half * 8;
        float* crow = C + (size_t)row * Nc + nBase + lm;
        crow[0]  = acc0[v];
        crow[16] = acc1[v];
        crow[32] = acc2[v];
        crow[48] = acc3[v];
    }
}

// Per-node attention logits: as[n,h] = dot(h[n,h,:], a_src[h,:]); same for dst.
template <int H, int C>
__global__ void alpha_kernel(const float* __restrict__ h, const float* __restrict__ a_src,
                             const float* __restrict__ a_dst,
                             float* __restrict__ as, float* __restrict__ ad, int n) {
    const int F = H * C;
    int node = blockIdx.x * (blockDim.x >> 5) + (threadIdx.x >> 5);
    if (node >= n) return;
    int lane = threadIdx.x & 31;
    float ps[H], pd[H];
#pragma unroll
    for (int hh = 0; hh < H; ++hh) { ps[hh] = 0.f; pd[hh] = 0.f; }
    const float* hp = h + (size_t)node * F;
    for (int c = lane; c < F; c += 32) {
        float hv = hp[c];
        int hh = c / C;
        ps[hh] += hv * a_src[c];
        pd[hh] += hv * a_dst[c];
    }
#pragma unroll
    for (int off = 16; off > 0; off >>= 1)
#pragma unroll
        for (int hh = 0; hh < H; ++hh) {
            ps[hh] += __shfl_xor(ps[hh], off, 32);
            pd[hh] += __shfl_xor(pd[hh], off, 32);
        }
    if (lane == 0)
#pragma unroll
        for (int hh = 0; hh < H; ++hh) {
            as[node * H + hh] = ps[hh];
            ad[node * H + hh] = pd[hh];
        }
}

// Pass 1: segment max over incoming edges (incl. self loops at e >= E).
template <int H>
__global__ void edge_max_kernel(const long long* __restrict__ ei,
                                const float* __restrict__ as, const float* __restrict__ ad,
                                float* __restrict__ mx, int E, int n) {
    int e = blockIdx.x * blockDim.x + threadIdx.x;
    if (e >= E + n) return;
    long long s, d;
    if (e < E) { s = ei[e]; d = ei[E + e]; } else { s = d = e - E; }
#pragma unroll
    for (int hh = 0; hh < H; ++hh) {
        float v = leaky(as[s * H + hh] + ad[d * H + hh]);
        atomicMaxF(&mx[d * H + hh], v);
    }
}

// Pass 2: segment sum of exp(e - max).
template <int H>
__global__ void edge_sum_kernel(const long long* __restrict__ ei,
                                const float* __restrict__ as, const float* __restrict__ ad,
                                const float* __restrict__ mx, float* __restrict__ dn,
                                int E, int n) {
    int e = blockIdx.x * blockDim.x + threadIdx.x;
    if (e >= E + n) return;
    long long s, d;
    if (e < E) { s = ei[e]; d = ei[E + e]; } else { s = d = e - E; }
#pragma unroll
    for (int hh = 0; hh < H; ++hh) {
        float v = leaky(as[s * H + hh] + ad[d * H + hh]);
        atomicAdd(&dn[d * H + hh], __expf(v - mx[d * H + hh]));
    }
}

// Pass 3: out[dst] += att * h[src]. Warp per edge; each lane owns F/32
// CONTIGUOUS channels (inside a single head since (F/32) | C), so each lane
// computes exactly one exp and its loads coalesce into b128 across the wave.
template <int H, int C>
__global__ void edge_aggr_kernel(const long long* __restrict__ ei,
                                 const float* __restrict__ as, const float* __restrict__ ad,
                                 const float* __restrict__ mx, const float* __restrict__ dn,
                                 const float* __restrict__ h, float* __restrict__ out,
                                 int E, int n) {
    const int F = H * C;
    const int P = F / 32;  // channels per lane (8 or 2)
    int e = blockIdx.x * (blockDim.x >> 5) + (threadIdx.x >> 5);
    if (e >= E + n) return;
    int lane = threadIdx.x & 31;
    long long s, d;
    if (e < E) { s = ei[e]; d = ei[E + e]; } else { s = d = e - E; }
    const float* hs = h + (size_t)s * F;
    float* od = out + (size_t)d * F;

    const int base = lane * P;
    const int hh   = base / C;          // this lane's head
    float v = leaky(as[s * H + hh] + ad[d * H + hh]);
    float a = __expf(v - mx[d * H + hh]) / (dn[d * H + hh] + 1e-16f);

    float vals[P];
#pragma unroll
    for (int j = 0; j < P; ++j) vals[j] = hs[base + j];
#pragma unroll
    for (int j = 0; j < P; ++j) atomicAdd(&od[base + j], vals[j] * a);
}

// x = elu(x + bias), bias length HIDC (power of two)
__global__ void elu_bias_kernel(float* __restrict__ x, const float* __restrict__ b, int n) {
    int i = blockIdx.x * blockDim.x + threadIdx.x;
    if (i >= n) return;
    float v = x[i] + b[i & (HIDC - 1)];
    x[i] = v > 0.f ? v : (__expf(v) - 1.f);
}

// out[i] += b2[i % 64] (layer-2 bias; heads==1 so mean() is identity)
__global__ void bias_out_kernel(float* __restrict__ out, const float* __restrict__ b, int n) {
    int i = blockIdx.x * blockDim.x + threadIdx.x;
    if (i >= n) return;
    out[i] = out[i] + b[i & (OUTC - 1)];
}

extern "C" void kernel_launch(void* const* d_in, const int* in_sizes, int n_in,
                              void* d_out, int out_size, void* d_ws, size_t ws_size,
                              hipStream_t stream) {
    const float*     x   = (const float*)d_in[0];
    const long long* ei  = (const long long*)d_in[1];   // int64 edge_index [2, E]
    const float*     W1  = (const float*)d_in[2];
    const float*     aS1 = (const float*)d_in[3];
    const float*     aD1 = (const float*)d_in[4];
    const float*     b1  = (const float*)d_in[5];
    const float*     W2  = (const float*)d_in[6];
    const float*     aS2 = (const float*)d_in[7];
    const float*     aD2 = (const float*)d_in[8];
    const float*     b2  = (const float*)d_in[9];
    float* out = (float*)d_out;

    const int N = NNODES, E = NEDGES;
    const int total = E + N;
    const int mTiles = N / 16;                // 3125
    const int gemmBlksX = (mTiles + 7) / 8;   // 8 waves (M tiles) per block

    // Workspace layout (floats): h1[N*256] (reused as h2[N*64]) | x2[N*256] | 4x alpha[N*4]
    float* ws  = (float*)d_ws;
    float* h1  = ws;
    float* x2  = h1 + (size_t)N * HIDC;
    float* as1 = x2 + (size_t)N * HIDC;
    float* ad1 = as1 + (size_t)N * HEADS1;
    float* mx1 = ad1 + (size_t)N * HEADS1;
    float* dn1 = mx1 + (size_t)N * HEADS1;
    float* h2  = h1;  // layer-1 features dead after aggregation

    const int TB = 256;
    // ---------------- Layer 1 ----------------
    fill_kernel<<<(N * HEADS1 + TB - 1) / TB, TB, 0, stream>>>(mx1, -INFINITY, N * HEADS1);
    fill_kernel<<<(N * HEADS1 + TB - 1) / TB, TB, 0, stream>>>(dn1, 0.f, N * HEADS1);
    fill_kernel<<<(N * HIDC + TB - 1) / TB, TB, 0, stream>>>(x2, 0.f, N * HIDC);

    gemm_wmma_f32<<<dim3(gemmBlksX, HIDC / 64), TB, 0, stream>>>(x, W1, h1, N, INC, HIDC);
    alpha_kernel<HEADS1, HIDC / HEADS1><<<(N + 7) / 8, TB, 0, stream>>>(h1, aS1, aD1, as1, ad1, N);
    edge_max_kernel<HEADS1><<<(total + TB - 1) / TB, TB, 0, stream>>>(ei, as1, ad1, mx1, E, N);
    edge_sum_kernel<HEADS1><<<(total + TB - 1) / TB, TB, 0, stream>>>(ei, as1, ad1, mx1, dn1, E, N);
    edge_aggr_kernel<HEADS1, HIDC / HEADS1><<<(total + 7) / 8, TB, 0, stream>>>(
        ei, as1, ad1, mx1, dn1, h1, x2, E, N);
    elu_bias_kernel<<<(N * HIDC + TB - 1) / TB, TB, 0, stream>>>(x2, b1, N * HIDC);

    // ---------------- Layer 2 (accumulate straight into d_out) ----------------
    fill_kernel<<<(N + TB - 1) / TB, TB, 0, stream>>>(mx1, -INFINITY, N);
    fill_kernel<<<(N + TB - 1) / TB, TB, 0, stream>>>(dn1, 0.f, N);
    fill_kernel<<<(N * OUTC + TB - 1) / TB, TB, 0, stream>>>(out, 0.f, N * OUTC);

    gemm_wmma_f32<<<dim3(gemmBlksX, OUTC / 64), TB, 0, stream>>>(x2, W2, h2, N, HIDC, OUTC);
    alpha_kernel<1, OUTC><<<(N + 7) / 8, TB, 0, stream>>>(h2, aS2, aD2, as1, ad1, N);
    edge_max_kernel<1><<<(total + TB - 1) / TB, TB, 0, stream>>>(ei, as1, ad1, mx1, E, N);
    edge_sum_kernel<1><<<(total + TB - 1) / TB, TB, 0, stream>>>(ei, as1, ad1, mx1, dn1, E, N);
    edge_aggr_kernel<1, OUTC><<<(total + 7) / 8, TB, 0, stream>>>(
        ei, as1, ad1, mx1, dn1, h2, out, E, N);
    bias_out_kernel<<<(N * OUTC + TB - 1) / TB, TB, 0, stream>>>(out, b2, N * OUTC);

    (void)in_sizes; (void)n_in; (void)out_size; (void)ws_size;
}